// TransitionLoss_85590108274924
// MI455X (gfx1250) — compile-verified
//
#include <hip/hip_runtime.h>
#include <hip/hip_bf16.h>
#include <math.h>

// Problem constants (match reference): B=256 samples, D=512 features, 32 classes.
#define NB   256
#define ND   512
#define MARGIN_1 0.3f
#define ALPHA_1  0.5f

typedef __attribute__((ext_vector_type(2))) float v2f;
typedef __attribute__((ext_vector_type(8))) float v8f;

// ---------------------------------------------------------------------------
// Kernel 1: sim = F * F^T  (256x256 fp32) via V_WMMA_F32_16X16X4_F32.
// One wave (32 threads) per 16x16 output tile; grid = 16x16 tiles.
// A-frag (16x4 f32, 2 VGPR): lane<16 -> (M=lane, K=k0..k0+1), lane>=16 -> K=k0+2..k0+3
// B-frag (4x16 f32, 2 VGPR): same striping over N; B[k][n] = F[n][k].
// C/D (16x16 f32, 8 VGPR): VGPR j -> row j (lanes 0-15) / row j+8 (lanes 16-31).
// ---------------------------------------------------------------------------
__global__ void __launch_bounds__(32)
sim_wmma_kernel(const float* __restrict__ F, float* __restrict__ sim) {
    const int tm   = blockIdx.x;          // row tile
    const int tn   = blockIdx.y;          // col tile
    const int lane = threadIdx.x;         // 0..31
    const int idx  = lane & 15;           // M for A, N for B
    const int kh   = (lane >> 4) * 2;     // 0 or 2: K sub-offset within the 4-chunk

    const float* arow = F + (size_t)(tm * 16 + idx) * ND;  // A row (M)
    const float* brow = F + (size_t)(tn * 16 + idx) * ND;  // B col source (N row of F)

    v8f c = {};
#pragma unroll 8
    for (int k = 0; k < ND; k += 4) {
        v2f a = *(const v2f*)(arow + k + kh);
        v2f b = *(const v2f*)(brow + k + kh);
        c = __builtin_amdgcn_wmma_f32_16x16x4_f32(
                /*neg_a=*/false, a, /*neg_b=*/false, b,
                /*c_mod=*/(short)0, c, /*reuse_a=*/false, /*reuse_b=*/false);
    }

    const int rbase = (lane >> 4) * 8;    // rows 0..7 or 8..15 of the tile
    const int col   = lane & 15;
#pragma unroll
    for (int j = 0; j < 8; ++j) {
        sim[(size_t)(tm * 16 + rbase + j) * NB + tn * 16 + col] = c[j];
    }
}

// ---------------------------------------------------------------------------
// Kernel 2: zero the scalar output (graph-capture safe; no memset needed).
// ---------------------------------------------------------------------------
__global__ void zero_out_kernel(float* out) {
    if (threadIdx.x == 0 && blockIdx.x == 0) out[0] = 0.0f;
}

// ---------------------------------------------------------------------------
// Kernel 3: per-anchor transition loss. One 256-thread workgroup per anchor.
// Categories per sample t (relative to anchor i):
//   cat = (pos?2:0) | (intra?1:0):  mnc=0, mni=1, mpc=2, mpi=3.
// Row t's softmax mask in the reference is always its *paired* category
// (same pos-ness, flipped sub-ness) = cat ^ 1, so each row needs exactly one
// (max, sum) stat pair.
// For each sign (neg: S=mni,K=mnc ; pos: S=mpi,K=mpc):
//   tr[r,c] = sum_{k in K} ab[r,k] * ba[k,c],  r,c in S
//   ab[r,k] = exp(sim[r,k]-m[r]) * inv_s[r],  ba[k,c] = exp(sim[k,c]-m[k]) * inv_s[k]
// ba is staged compacted in LDS (f16, |K|*|S| <= 128^2 = 16384 elems = 32 KB),
// then rows of ab are produced cooperatively; thread t owns output column t.
// ---------------------------------------------------------------------------
__global__ void __launch_bounds__(NB)
transition_loss_kernel(const float* __restrict__ sim,
                       const int*   __restrict__ sub,
                       const int*   __restrict__ targets,
                       float* __restrict__ out) {
    __shared__ int      cat[NB];
    __shared__ float    mrow[NB];
    __shared__ float    sinv[NB];
    __shared__ int      listS[NB];
    __shared__ int      listK[NB];
    __shared__ int      nS_sh, nK_sh;
    __shared__ _Float16 eba[128 * 128];   // |K| x |S| compacted ba, 32 KB
    __shared__ float    abrow[NB];
    __shared__ float    red[NB];

    const int i = blockIdx.x;   // anchor
    const int t = threadIdx.x;  // sample index / worker

    const int ti = targets[i];
    const int si = sub[i];
    {
        const bool pos  = (targets[t] == ti);
        const bool intr = (sub[t] == si);
        cat[t] = (pos ? 2 : 0) | (intr ? 1 : 0);
    }
    __syncthreads();

    // ---- per-row softmax stats vs paired mask (cat ^ 1) ----
    {
        const int pm = cat[t] ^ 1;
        const float* srow = sim + (size_t)t * NB;
        float m = -INFINITY;
        for (int k = 0; k < NB; ++k)
            if (cat[k] == pm) m = fmaxf(m, srow[k]);
        if (!(m > -INFINITY)) m = 0.0f;           // empty mask -> 0 (matches ref)
        float s = 0.0f;
        for (int k = 0; k < NB; ++k)
            if (cat[k] == pm) s += __expf(srow[k] - m);
        mrow[t] = m;
        sinv[t] = 1.0f / fmaxf(s, 1e-30f);
    }
    __syncthreads();

    float anchor_loss = 0.0f;  // meaningful on thread 0 only

    for (int sign = 0; sign < 2; ++sign) {
        const int Scat = sign ? 3 : 1;   // mpi : mni
        const int Kcat = sign ? 2 : 0;   // mpc : mnc

        if (t == 0) {
            int ns = 0, nk = 0;
            for (int k = 0; k < NB; ++k) {
                if (cat[k] == Scat)      listS[ns++] = k;
                else if (cat[k] == Kcat) listK[nk++] = k;
            }
            nS_sh = ns; nK_sh = nk;
        }
        __syncthreads();
        const int nS = nS_sh;
        const int nK = nK_sh;

        // stage ba (compacted, normalized) into LDS:  eba[kIdx*nS + cIdx]
        for (int idx = t; idx < nS * nK; idx += NB) {
            const int kIdx = idx / nS;
            const int cIdx = idx - kIdx * nS;
            const int k = listK[kIdx];
            const int cc = listS[cIdx];
            eba[idx] = (_Float16)(__expf(sim[(size_t)k * NB + cc] - mrow[k]) * sinv[k]);
        }
        __syncthreads();

        float acc_sum = 0.0f;
        for (int rIdx = 0; rIdx < nS; ++rIdx) {
            const int r = listS[rIdx];
            // cooperative ab row (un-normalized; sinv[r] folded in at the end)
            for (int kIdx = t; kIdx < nK; kIdx += NB)
                abrow[kIdx] = __expf(sim[(size_t)r * NB + listK[kIdx]] - mrow[r]);
            __syncthreads();

            if (t < nS) {  // thread t owns output column cIdx = t
                float acc = 0.0f;
                for (int kIdx = 0; kIdx < nK; ++kIdx)
                    acc += abrow[kIdx] * (float)eba[kIdx * nS + t];
                const float tr = acc * sinv[r];
                const float tv = __builtin_sqrtf(fmaxf(tr, 1e-12f));
                // neg: max(0, sqrt + (margin - alpha)) ; pos: max(0, alpha - sqrt)
                acc_sum += sign ? fmaxf(ALPHA_1 - tv, 0.0f)
                                : fmaxf(tv + (MARGIN_1 - ALPHA_1), 0.0f);
            }
            __syncthreads();
        }

        // block reduction of acc_sum
        red[t] = acc_sum;
        __syncthreads();
        for (int off = NB / 2; off > 0; off >>= 1) {
            if (t < off) red[t] += red[t + off];
            __syncthreads();
        }
        if (t == 0) {
            const float cnt = (float)((nS * nS) > 0 ? (nS * nS) : 1);
            anchor_loss += red[0] / cnt;
        }
        __syncthreads();
    }

    if (t == 0) atomicAdd(out, anchor_loss * (1.0f / (float)NB));
}

// ---------------------------------------------------------------------------
// Launch: zero out -> sim GEMM (WMMA fp32) -> per-anchor loss.
// Workspace layout: sim matrix at d_ws (256*256 fp32 = 256 KB).
// ---------------------------------------------------------------------------
extern "C" void kernel_launch(void* const* d_in, const int* in_sizes, int n_in,
                              void* d_out, int out_size, void* d_ws, size_t ws_size,
                              hipStream_t stream) {
    const float* feature = (const float*)d_in[0];   // [256, 512] f32
    const int*   sub     = (const int*)d_in[1];     // [256]
    const int*   targets = (const int*)d_in[2];     // [256]
    float*       out     = (float*)d_out;           // scalar f32
    float*       sim     = (float*)d_ws;            // 256x256 f32 scratch

    zero_out_kernel<<<1, 32, 0, stream>>>(out);

    dim3 gsim(NB / 16, NB / 16);   // 16x16 tiles
    sim_wmma_kernel<<<gsim, 32, 0, stream>>>(feature, sim);

    transition_loss_kernel<<<NB, NB, 0, stream>>>(sim, sub, targets, out);
}